// CEMA_66340064854504
// MI455X (gfx1250) — compile-verified
//
#include <hip/hip_runtime.h>

// CEMA (diagonal linear recurrence) as chunked causal matmuls on CDNA5 WMMA
// (v_wmma_f32_16x16x4_f32), gfx1250 / wave32, async-to-LDS staging of x,
// and pre-swizzled operand fragments in LDS (immediate-offset ds_load_b64).
//
//   y[b,t,e]   = sum_{k<=t} K_e(t-k) x[b,k,e],  K_e(d) = sum_n g*p*q^d
//   chunk L=64: y_part = T_e @ x_chunk  (T lower-triangular Toeplitz of K)
//               h_part = P_e @ x_chunk  (P[n,j] = p*q^(L-1-j))
//   carry scan: h_in(c+1) = q^L (*) h_in(c) + h_part(c)
//   fixup:      y[t0+i]  += U_e[i,:] . h_in(c),  U[i,n] = g*q^(i+1)

typedef float v2f __attribute__((ext_vector_type(2)));
typedef float v8f __attribute__((ext_vector_type(8)));

#define B_ 8
#define S_ 4096
#define E_ 512
#define N_ 16
#define L_ 64           // chunk length (K and M of the per-e GEMM)
#define C_ (S_ / L_)    // 64 chunks per sequence
#define XPAD 66         // padded column stride (floats): bank-conflict free,
                        // 264 B = 0 mod 8 so even-k b64 loads stay aligned

__device__ __forceinline__ float qpow(float q, int e) {
    float r = 1.0f, b = q;
    while (e) { if (e & 1) r *= b; b *= b; e >>= 1; }
    return r;
}

// ---------------------------------------------------------------------------
// Kernel 1: per-chunk partial outputs (y_part) and chunk state contributions
// (h_part) via FP32 WMMA.  One block = 4 waves, all on the same e; each wave
// owns a 16-column tile where a column is one (batch, chunk) pair.
// grid = E * 8 blocks, block = 128 threads.
// ---------------------------------------------------------------------------
__global__ __launch_bounds__(128)
void cema_partial_wmma(const float* __restrict__ x,
                       const float* __restrict__ p_coeff,
                       const float* __restrict__ q_coeff,
                       const float* __restrict__ gmat,
                       float* __restrict__ out,
                       float* __restrict__ hpart) {
    // Kext[64+d] = K_e(d), Kext[0..63] = 0 (zero pad kills the causal branch)
    __shared__ float Kext[2 * L_];
    // Pre-swizzled WMMA A-fragments, exactly in register order:
    //   tiles 0..3 = T_e row-tiles, tile 4 = P_e.  One float2 per lane.
    __shared__ v2f  Apre[L_ / 4][5][32];
    __shared__ float Xt[4][16][XPAD];       // per-wave X tile: [col][k]

    const int e   = blockIdx.x >> 3;
    const int tid = threadIdx.x;

    const int wave = tid >> 5;
    const int lane = tid & 31;
    const int half = lane >> 4;             // 0: lanes 0-15, 1: lanes 16-31
    const int m    = lane & 15;

    const int ct  = ((blockIdx.x & 7) << 2) + wave;  // column tile 0..31
    const int col = ct * 16 + m;                     // 0..511 = b*64 + chunk
    const int b   = col >> 6;
    const int c   = col & (C_ - 1);

    // ---- kick off async DMA of this wave's 64x16 X tile into LDS ----
    // lane layout: column = lane&15, k-half = lane>>4 (k = half*32 + j)
    {
        const int    sb = half * 32;        // 0 or 32
        const float* gp = x + ((size_t)b * S_ + (size_t)c * L_ + sb) * E_ + e;
        unsigned long long gaddr = (unsigned long long)(uintptr_t)gp;
        unsigned int laddr = (unsigned int)(uintptr_t)&Xt[wave][m][sb];
#pragma unroll
        for (int j = 0; j < 32; ++j) {
            asm volatile("global_load_async_to_lds_b32 %0, %1, off"
                         :: "v"(laddr), "v"(gaddr) : "memory");
            gaddr += (unsigned long long)E_ * 4ull;
            laddr += 4u;
        }
    }

    // ---- build kernel tables while the async engine fills Xt ----
    {
        const int d = tid - L_;             // tid 0..63 -> zero pad
        float s = 0.0f;
        if (d >= 0) {
#pragma unroll
            for (int n = 0; n < N_; ++n) {
                const float p = p_coeff[e * N_ + n];
                const float q = q_coeff[e * N_ + n];
                const float g = gmat[e * N_ + n];
                s += g * p * qpow(q, d);
            }
        }
        Kext[tid] = s;
    }
    __syncthreads();                        // Kext ready

    // Toeplitz tiles: Apre[kk][t][l] = { K(i-k), K(i-k-1) }, i = 16t + m
    for (int idx = tid; idx < (L_ / 4) * 4 * 32; idx += 128) {
        const int kk  = idx >> 7;
        const int rem = idx & 127;
        const int t   = rem >> 5;
        const int l   = rem & 31;
        const int k   = kk * 4 + 2 * (l >> 4);
        const int i0  = L_ + 16 * t + (l & 15) - k;
        v2f v;
        v.x = Kext[i0];
        v.y = Kext[i0 - 1];
        Apre[kk][t][l] = v;
    }
    // P tile: Apre[kk][4][l] = { p*q^(63-k), p*q^(62-k) }, row n = l&15
    for (int idx = tid; idx < (L_ / 4) * 32; idx += 128) {
        const int kk = idx >> 5;
        const int l  = idx & 31;
        const int k  = kk * 4 + 2 * (l >> 4);
        const float p = p_coeff[e * N_ + (l & 15)];
        const float q = q_coeff[e * N_ + (l & 15)];
        v2f v;
        v.x = p * qpow(q, L_ - 1 - k);
        v.y = p * qpow(q, L_ - 2 - k);
        Apre[kk][4][l] = v;
    }

    asm volatile("s_wait_asynccnt 0x0" ::: "memory");
    __syncthreads();                        // Apre + Xt visible to all waves

    v8f acc0 = {}, acc1 = {}, acc2 = {}, acc3 = {};  // y rows 0-15/.../48-63
    v8f hacc = {};                                   // h_part rows n=0..15

    // All LDS reads below: single base VGPR + immediate DS offset, aligned b64.
    const v2f* __restrict__ abase = &Apre[0][0][lane];         // stride 32 v2f
    const v2f* __restrict__ bbase = (const v2f*)&Xt[wave][m][2 * half];

#pragma unroll
    for (int kk = 0; kk < L_ / 4; ++kk) {
        const v2f bv = bbase[kk * 2];          // X[k..k+1][col], k = 4kk+2*half
        const v2f a0 = abase[(kk * 5 + 0) * 32];
        const v2f a1 = abase[(kk * 5 + 1) * 32];
        const v2f a2 = abase[(kk * 5 + 2) * 32];
        const v2f a3 = abase[(kk * 5 + 3) * 32];
        const v2f pv = abase[(kk * 5 + 4) * 32];

        acc0 = __builtin_amdgcn_wmma_f32_16x16x4_f32(false, a0, false, bv,
                                                     (short)0, acc0, false, false);
        acc1 = __builtin_amdgcn_wmma_f32_16x16x4_f32(false, a1, false, bv,
                                                     (short)0, acc1, false, false);
        acc2 = __builtin_amdgcn_wmma_f32_16x16x4_f32(false, a2, false, bv,
                                                     (short)0, acc2, false, false);
        acc3 = __builtin_amdgcn_wmma_f32_16x16x4_f32(false, a3, false, bv,
                                                     (short)0, acc3, false, false);
        hacc = __builtin_amdgcn_wmma_f32_16x16x4_f32(false, pv, false, bv,
                                                     (short)0, hacc, false, false);
    }

    // D layout: VGPR r -> row (r + 8*half) of the 16-row tile, column m.
    float* ocol = out + ((size_t)b * S_ + (size_t)c * L_) * E_ + e;
#pragma unroll
    for (int r = 0; r < 8; ++r) {
        const int rr = r + 8 * half;
        ocol[(size_t)(rr)      * E_] = acc0[r];
        ocol[(size_t)(rr + 16) * E_] = acc1[r];
        ocol[(size_t)(rr + 32) * E_] = acc2[r];
        ocol[(size_t)(rr + 48) * E_] = acc3[r];
    }
    float* hrow = hpart + (((size_t)b * E_ + e) * C_ + c) * N_;
#pragma unroll
    for (int r = 0; r < 8; ++r) hrow[r + 8 * half] = hacc[r];
}

// ---------------------------------------------------------------------------
// Kernel 2: sequential carry over the 64 chunks of each (b,e) channel plus
// the cross-chunk output correction  y[t0+i] += U[i,:] . h_in.
// One wave per (b,e): lane n<16 carries h[n]; every lane fixes up rows
// i = lane and i = lane+32 of each chunk.  grid = B*E, block = 32.
// ---------------------------------------------------------------------------
__global__ __launch_bounds__(32)
void cema_carry_fixup(const float* __restrict__ q_coeff,
                      const float* __restrict__ gmat,
                      const float* __restrict__ hpart,
                      float* __restrict__ out) {
    const int bid  = blockIdx.x;
    const int b    = bid / E_;
    const int e    = bid % E_;
    const int lane = threadIdx.x;

    float u1[N_], u2[N_];
#pragma unroll
    for (int n = 0; n < N_; ++n) {
        const float q = q_coeff[e * N_ + n];
        const float g = gmat[e * N_ + n];
        u1[n] = g * qpow(q, lane + 1);    // U[lane][n]
        u2[n] = g * qpow(q, lane + 33);   // U[lane+32][n]
    }
    const float qLv = (lane < N_) ? qpow(q_coeff[e * N_ + lane], L_) : 0.0f;

    float h = 0.0f;  // h_in for this lane's n (lanes >= 16 carry zeros)
    const float* hrow = hpart + ((size_t)b * E_ + e) * C_ * N_;
    float* ob = out + (size_t)b * S_ * E_ + e;

    for (int c = 0; c < C_; ++c) {
        float corr1 = 0.0f, corr2 = 0.0f;
#pragma unroll
        for (int n = 0; n < N_; ++n) {
            const float hb = __shfl(h, n, 32);
            corr1 += u1[n] * hb;
            corr2 += u2[n] * hb;
        }
        const size_t t1 = (size_t)c * L_ + lane;
        ob[t1 * E_]        += corr1;
        ob[(t1 + 32) * E_] += corr2;

        const float hp = (lane < N_) ? hrow[c * N_ + lane] : 0.0f;
        h = qLv * h + hp;
    }
}

// ---------------------------------------------------------------------------
extern "C" void kernel_launch(void* const* d_in, const int* in_sizes, int n_in,
                              void* d_out, int out_size, void* d_ws, size_t ws_size,
                              hipStream_t stream) {
    (void)in_sizes; (void)n_in; (void)out_size; (void)ws_size;

    const float* x = (const float*)d_in[0];
    // d_in[1] = omega: unused by the reference's returned math.
    const float* p = (const float*)d_in[2];
    const float* q = (const float*)d_in[3];
    const float* g = (const float*)d_in[4];   // gamma[E,N,1] -> flat [E,N]

    float* out   = (float*)d_out;
    float* hpart = (float*)d_ws;  // needs B*E*C*N*4 = 16.8 MB of scratch

    cema_partial_wmma<<<E_ * 8, 128, 0, stream>>>(x, p, q, g, out, hpart);
    cema_carry_fixup<<<B_ * E_, 32, 0, stream>>>(q, g, hpart, out);
}